// NeuralNetwork_56985626083963
// MI455X (gfx1250) — compile-verified
//
#include <hip/hip_runtime.h>
#include <math.h>

// ---------------------------------------------------------------------------
// 7-layer fp32 MLP chain (1024-wide), one kernel launch per layer.
// GEMV expressed as V_WMMA_F32_16X16X4_F32 with the activation vector
// replicated across all 16 B-columns (every D column = W_tile @ h).
// Memory-bound: 7 x 4MB fp32 weights (~29 MB, L2-resident on replay).
// ---------------------------------------------------------------------------

typedef __attribute__((ext_vector_type(2))) float v2f;
typedef __attribute__((ext_vector_type(8))) float v8f;

#define MLP_WIDTH 1024
#define MLP_NROW  8192              // N = N_LAYERS * WIDTH
#define MLP_STRIDE 8193             // parameter_matrix row stride (N+1)
#define NWAVE 8                     // waves per block (wave32)
#define KCHUNK (MLP_WIDTH / NWAVE)  // 128 K-elements per wave

template <bool ACT>
__global__ __launch_bounds__(256)
void mlp_layer(const float* __restrict__ P,     // parameter_matrix (8192 x 8193)
               const float* __restrict__ hin,   // input activations (1024)
               float* __restrict__ hout,        // output activations (1024)
               int layer)                       // 1..7
{
    __shared__ __align__(16) float sh[MLP_WIDTH];   // staged h
    __shared__ float part[NWAVE][16];               // per-wave partial outputs

    const int tid  = threadIdx.x;
    const int wave = tid >> 5;
    const int lane = tid & 31;
    const int tile = blockIdx.x;                    // 0..63 -> 16 output rows each

    const size_t rowBase = (size_t)layer * MLP_WIDTH + (size_t)tile * 16;
    const size_t colBase = (size_t)(layer - 1) * MLP_WIDTH;

    // Cooperatively stage h into LDS: 256 threads x 4 floats = 1024 floats.
    reinterpret_cast<float4*>(sh)[tid] =
        reinterpret_cast<const float4*>(hin)[tid];
    __syncthreads();

    // WMMA A-matrix layout (f32 16x4): lane holds row M = lane&15,
    // K slots (lane>>4)*2 + {0,1}.  B (4x16) uses the same K->slot mapping
    // since all 16 columns are replicated copies of the h chunk.
    const int row  = lane & 15;
    const int half = lane >> 4;

    const float* __restrict__ arow =
        P + (rowBase + (size_t)row) * (size_t)MLP_STRIDE + colBase
          + (size_t)(wave * KCHUNK);
    const float* __restrict__ brow = sh + wave * KCHUNK;

    v8f acc = {};
#pragma unroll 8
    for (int kk = 0; kk < KCHUNK; kk += 4) {
        const int o = kk + half * 2;
        // One near-scope prefetch per 8-iteration group, 256B ahead in-row.
        if ((kk & 28) == 0)
            __builtin_prefetch(arow + o + 64, /*rw=*/0, /*locality=*/3);
        v2f a, b;
        a.x = arow[o];
        a.y = arow[o + 1];
        b.x = brow[o];
        b.y = brow[o + 1];
        // D = A(16x4) * B(4x16) + C ;  fp32-exact accumulation path
        acc = __builtin_amdgcn_wmma_f32_16x16x4_f32(
            /*neg_a=*/false, a, /*neg_b=*/false, b,
            /*c_mod=*/(short)0, acc, /*reuse_a=*/false, /*reuse_b=*/false);
    }

    // D layout: lane 0 carries M=0..7 in acc[0..7], lane 16 carries M=8..15.
    if (lane == 0) {
#pragma unroll
        for (int i = 0; i < 8; ++i) part[wave][i] = acc[i];
    } else if (lane == 16) {
#pragma unroll
        for (int i = 0; i < 8; ++i) part[wave][8 + i] = acc[i];
    }
    __syncthreads();

    // Combine the 8 K-slice partials, add bias, activation, store.
    if (tid < 16) {
        float s = 0.f;
#pragma unroll
        for (int w = 0; w < NWAVE; ++w) s += part[w][tid];
        s += P[(rowBase + (size_t)tid) * (size_t)MLP_STRIDE + MLP_NROW]; // bias
        if (ACT) s = s / (1.0f + __expf(-s));    // SiLU
        hout[tile * 16 + tid] = s;
    }
}

extern "C" void kernel_launch(void* const* d_in, const int* in_sizes, int n_in,
                              void* d_out, int out_size, void* d_ws, size_t ws_size,
                              hipStream_t stream)
{
    (void)in_sizes; (void)n_in; (void)out_size; (void)ws_size;

    const float* x = (const float*)d_in[0];          // (1024,)
    const float* P = (const float*)d_in[1];          // (8192, 8193)
    float* out = (float*)d_out;                      // (1024,)
    float* buf = (float*)d_ws;                       // 2 x 1024 f32 ping-pong

    // out(l): l=1->buf0, l=2->buf1, l=3->buf0, ... ; in(l>=2) = out(l-1)
    for (int l = 1; l <= 7; ++l) {
        const float* hin  = (l == 1) ? x   : (buf + (size_t)(l & 1) * MLP_WIDTH);
        float*       hout = (l == 7) ? out : (buf + (size_t)((l + 1) & 1) * MLP_WIDTH);
        if (l < 7)
            mlp_layer<true><<<dim3(64), dim3(256), 0, stream>>>(P, hin, hout, l);
        else
            mlp_layer<false><<<dim3(64), dim3(256), 0, stream>>>(P, hin, hout, l);
    }
}